// ElectrostaticEnergyBlock_343597384389
// MI455X (gfx1250) — compile-verified
//
#include <hip/hip_runtime.h>

// ---------------------------------------------------------------------------
// ElectrostaticEnergyBlock for MI455X (gfx1250, wave32)
// Memory-bound edge kernel: NT-streamed int64 edge list (102 MB, read once),
// cached gathers over 2.8 MB node tables (fits L2/WGP$), f32 HW atomics for
// the segment sums. All math is branch-free: TRANS ops (v_rsq/v_rcp/v_exp)
// replace IEEE div/sqrt chains and the divergent libm erff.
// NOTE: ee_edge_kernel is defined FIRST so the disasm snippet shows it.
// ---------------------------------------------------------------------------

typedef long long i64;
typedef i64 v2i64 __attribute__((ext_vector_type(2)));   // 16B index-pair load

struct F3 { float x, y, z; };                            // 12B node rows

// Physical constants, folded at compile time in double precision.
constexpr double PI_D   = 3.14159265358979323846;
constexpr double EPS0_D = 8.8541878128e-12;
constexpr double C_D    = 299792458.0;
constexpr double ECH_D  = 1.602176634e-19;
constexpr double KE_D   = 1.0 / (4.0 * PI_D * EPS0_D);   // Coulomb constant

// Hyperparameters: QQ_SIGMA=1.4, MU_LAMBDA=0.2, MU_ALPHA=6.0
constexpr float ALPHA_QQ  = (float)(1.0 / 2.8);                      // 1/sqrt(4*sigma^2)
constexpr float ALPHA2_QQ = (float)(1.0 / (2.8 * 2.8));              // alpha^2
constexpr float C_QQ      = (float)(0.5 * KE_D * ECH_D * 1e10);      // E_qq prefactor
constexpr float SOFT_ADD  = (float)(6.0 * 0.8 * 0.8);                // alpha*(1-lambda)^2
constexpr float C_QMU     = (float)(KE_D * 0.1 / C_D);               // E_q_mu prefactor
constexpr float C_MUMU    = (float)(0.5 * KE_D * 1e-12 / (C_D * C_D) / ECH_D);

// Abramowitz & Stegun 7.1.26 erf coefficients (max abs err 1.5e-7, x >= 0).
constexpr float ERF_P  = 0.3275911f;
constexpr float ERF_A1 = 0.254829592f;
constexpr float ERF_A2 = -0.284496736f;
constexpr float ERF_A3 = 1.421413741f;
constexpr float ERF_A4 = -1.453152027f;
constexpr float ERF_A5 = 1.061405429f;

// Branch-free erf for x >= 0; 'nx2' is -x^2 (avoids re-squaring).
// NaN in -> NaN out (matches reference's NaN on zero-length edges).
__device__ __forceinline__ float ee_erf_pos(float x, float nx2) {
    const float t    = __builtin_amdgcn_rcpf(fmaf(ERF_P, x, 1.0f));  // v_rcp_f32
    const float expv = __expf(nx2);                                  // v_exp_f32
    float p = fmaf(ERF_A5, t, ERF_A4);
    p = fmaf(p, t, ERF_A3);
    p = fmaf(p, t, ERF_A2);
    p = fmaf(p, t, ERF_A1);
    p = p * t;
    return fmaf(-p, expv, 1.0f);
}

// One edge's full computation + scatter. Inlined into the pair loop.
__device__ __forceinline__
void ee_process_edge(int s, int r,
                     const float* __restrict__ charge,
                     const F3*   __restrict__ pos,
                     const F3*   __restrict__ dip,
                     float* __restrict__ Eq,
                     float* __restrict__ Emu) {
    // Cached gathers (2.8 MB footprint -> L2/WGP$ hits)
    const F3 ps = pos[s];
    const F3 pr = pos[r];
    const F3 ms = dip[s];
    const F3 mr = dip[r];
    const float qs = charge[s];
    const float qr = charge[r];

    const float Rx = ps.x - pr.x;
    const float Ry = ps.y - pr.y;
    const float Rz = ps.z - pr.z;
    const float d2 = fmaf(Rx, Rx, fmaf(Ry, Ry, Rz * Rz));

    const float R_mu_j = fmaf(Rx, ms.x, fmaf(Ry, ms.y, Rz * ms.z));
    const float R_mu_i = fmaf(Rx, mr.x, fmaf(Ry, mr.y, Rz * mr.z));  // = -mu_i_R_ij
    const float mu_mu  = fmaf(mr.x, ms.x, fmaf(mr.y, ms.y, mr.z * ms.z));
    const float qq     = qr * qs;

    // E_Gauss_qq: 1/d and d from one v_rsq_f32; erf via branch-free rational.
    const float rinv_d = __builtin_amdgcn_rsqf(d2);            // 1/d
    const float d      = d2 * rinv_d;                          // d (NaN at d2==0)
    const float erfv   = ee_erf_pos(ALPHA_QQ * d, -ALPHA2_QQ * d2);
    const float E_qq   = C_QQ * (qq * rinv_d) * erfv;

    // Softened dipole terms (reference passes d^2 as its "d_ij").
    // soft >= 3.84 always -> raw rsq is safe: inv = rsq^2, inv^1.5 = rsq^3.
    const float soft  = d2 + SOFT_ADD;
    const float rsq   = __builtin_amdgcn_rsqf(soft);           // soft^-0.5
    const float inv   = rsq * rsq;                             // soft^-1
    const float inv15 = inv * rsq;                             // soft^-1.5

    const float E_qmu  = C_QMU * R_mu_j * qr * inv15;
    // term2 = mu_mu - 3*R_mu_j*(-R_mu_i)*inv = mu_mu + 3*R_mu_j*R_mu_i*inv
    const float term2  = fmaf(3.0f * R_mu_j * R_mu_i, inv, mu_mu);
    const float E_mumu = C_MUMU * inv15 * term2;

    // Segment-sum via HW f32 atomics (no-return -> STOREcnt path)
    unsafeAtomicAdd(&Eq[r],  E_qq);
    unsafeAtomicAdd(&Emu[r], E_qmu + E_mumu);
}

// --- HOT KERNEL (defined first so the asm snippet shows its body) ---------
__global__ __launch_bounds__(256)
void ee_edge_kernel(const float* __restrict__ charge,     // [N]
                    const float* __restrict__ dipoles,    // [N,3]
                    const float* __restrict__ positions,  // [N,3]
                    const i64*  __restrict__ sender,      // [E]
                    const i64*  __restrict__ receiver,    // [E]
                    float* __restrict__ Eq,               // [N] (atomic accum)
                    float* __restrict__ Emu,              // [N] (atomic accum)
                    int pairs) {
    const int t = blockIdx.x * blockDim.x + threadIdx.x;
    if (t >= pairs) return;

    const F3* pos = reinterpret_cast<const F3*>(positions);
    const F3* dip = reinterpret_cast<const F3*>(dipoles);

    // Streaming index read: non-temporal b128 (2 edges per thread).
    // NT keeps the one-shot 102 MB stream from evicting hot node tables.
    const v2i64* sp_ptr = reinterpret_cast<const v2i64*>(sender)   + t;
    const v2i64* rp_ptr = reinterpret_cast<const v2i64*>(receiver) + t;
    const v2i64 sp = __builtin_nontemporal_load(sp_ptr);
    const v2i64 rp = __builtin_nontemporal_load(rp_ptr);

    // Run the index stream ahead of the launch wave (global_prefetch_b8).
    const int PF = 16384;   // threads ahead (~256 KB per array)
    if (t + PF < pairs) {
        __builtin_prefetch(sp_ptr + PF, 0, 0);
        __builtin_prefetch(rp_ptr + PF, 0, 0);
    }

    ee_process_edge((int)sp[0], (int)rp[0], charge, pos, dip, Eq, Emu);
    ee_process_edge((int)sp[1], (int)rp[1], charge, pos, dip, Eq, Emu);
}

__global__ void ee_zero_kernel(float* __restrict__ out, int n) {
    int i = blockIdx.x * blockDim.x + threadIdx.x;
    if (i < n) out[i] = 0.0f;
}

// Odd-E tail: one edge, launched (one 32-thread block, thread 0 works) only when n_edges is odd.
__global__ void ee_edge_tail_kernel(const float* __restrict__ charge,
                                    const float* __restrict__ dipoles,
                                    const float* __restrict__ positions,
                                    const i64*  __restrict__ sender,
                                    const i64*  __restrict__ receiver,
                                    float* __restrict__ Eq,
                                    float* __restrict__ Emu,
                                    int e) {
    if (threadIdx.x == 0 && blockIdx.x == 0) {
        const F3* pos = reinterpret_cast<const F3*>(positions);
        const F3* dip = reinterpret_cast<const F3*>(dipoles);
        ee_process_edge((int)sender[e], (int)receiver[e], charge, pos, dip, Eq, Emu);
    }
}

extern "C" void kernel_launch(void* const* d_in, const int* in_sizes, int n_in,
                              void* d_out, int out_size, void* d_ws, size_t ws_size,
                              hipStream_t stream) {
    (void)n_in; (void)d_ws; (void)ws_size; (void)out_size;

    const float* charge    = (const float*)d_in[0];   // [N]
    const float* dipoles   = (const float*)d_in[1];   // [N,3]
    const float* positions = (const float*)d_in[2];   // [N,3]
    const i64*   edge_idx  = (const i64*)d_in[3];     // [2,E] int64

    const int n_nodes = in_sizes[0];
    const int n_edges = in_sizes[3] / 2;

    float* Eq  = (float*)d_out;                       // output 0: E_q  [N]
    float* Emu = Eq + n_nodes;                        // output 1: E_mu [N]

    const i64* sender   = edge_idx;
    const i64* receiver = edge_idx + n_edges;

    // Zero the accumulators (harness poisons d_out; atomics accumulate).
    const int nz = 2 * n_nodes;
    ee_zero_kernel<<<(nz + 255) / 256, 256, 0, stream>>>((float*)d_out, nz);

    // Main kernel: 2 edges per thread, branch-free body.
    const int pairs  = n_edges >> 1;
    const int blocks = (pairs + 255) / 256;
    ee_edge_kernel<<<blocks, 256, 0, stream>>>(charge, dipoles, positions,
                                               sender, receiver, Eq, Emu, pairs);

    // Tail edge only if E is odd (size-dependent, deterministic; E=6.4M -> never).
    if (n_edges & 1) {
        ee_edge_tail_kernel<<<1, 32, 0, stream>>>(charge, dipoles, positions,
                                                  sender, receiver, Eq, Emu,
                                                  n_edges - 1);
    }
}